// ScaledDotProductAttention_83150566851078
// MI455X (gfx1250) — compile-verified
//
#include <hip/hip_runtime.h>
#include <math.h>

typedef __attribute__((ext_vector_type(2))) float v2f;
typedef __attribute__((ext_vector_type(8))) float v8f;

#define S_LEN 1024
#define D_DIM 64

// D = A(16x4) * B(4x16) + C, full fp32 (V_WMMA_F32_16X16X4_F32)
__device__ __forceinline__ v8f wmma4(v2f a, v2f b, v8f c) {
  return __builtin_amdgcn_wmma_f32_16x16x4_f32(false, a, false, b, (short)0, c,
                                               false, false);
}

// Process one 16-key tile. MASKED=false is the steady-state path (all 16 key
// rows valid -> no cndmask selects at all); MASKED=true handles the single
// partial tile that straddles `len`.
template <bool MASKED>
__device__ __forceinline__ void process_tile(const float* __restrict__ Kb,
                                             const float* __restrict__ Vb,
                                             const v2f (&qf)[16], v8f (&acc)[4],
                                             float& mrow, float& lrow,
                                             int kbase, int len, int m, int h) {
  const v2f zero2 = {0.f, 0.f};

  // ---- Preload ALL K-tile fragments: issues as one big load clause ----
  v2f kf[16];
  {
    const float* kp = Kb + (size_t)(kbase + m) * D_DIM + 2 * h;
#pragma unroll
    for (int kc = 0; kc < 16; ++kc) kf[kc] = *(const v2f*)(kp + 4 * kc);
  }
  // ---- Preload ALL V-tile fragments (A-operand of ctx^T matmul) ----
  // vf[kc][j]: rows kbase+4kc+2h+{0,1}, column 16j+m (coalesced across lanes)
  v2f vf[4][4];
#pragma unroll
  for (int kc = 0; kc < 4; ++kc) {
    const float* vp0 = Vb + (size_t)(kbase + 4 * kc + 2 * h) * D_DIM + m;
    const float* vp1 = vp0 + D_DIM;
#pragma unroll
    for (int j = 0; j < 4; ++j) {
      vf[kc][j].x = vp0[16 * j];
      vf[kc][j].y = vp1[16 * j];
    }
  }

  // ---- S^T tile = K_tile(16x64) x Q^T(64x16), two parallel WMMA chains ----
  v8f sa, sb;
#pragma unroll
  for (int v = 0; v < 8; ++v) { sa[v] = 0.f; sb[v] = 0.f; }
  {
    const bool kv = MASKED ? ((kbase + m) < len) : true;
#pragma unroll
    for (int kc = 0; kc < 16; kc += 2) {
      v2f a0 = MASKED ? (kv ? kf[kc] : zero2) : kf[kc];
      v2f a1 = MASKED ? (kv ? kf[kc + 1] : zero2) : kf[kc + 1];
      sa = wmma4(a0, qf[kc], sa);
      sb = wmma4(a1, qf[kc + 1], sb);
    }
  }
  v8f st;
  // masked_fill(scores==0, 1e-10) BEFORE the 1/sqrt(64) scale (ref semantics)
#pragma unroll
  for (int v = 0; v < 8; ++v) {
    float s = sa[v] + sb[v];
    s = (s == 0.0f) ? 1e-10f : s;
    st[v] = s * 0.125f;
  }

  // ---- online softmax over keys; query index = lane, so state is per-lane ----
  float rmax = st[0];
#pragma unroll
  for (int v = 1; v < 8; ++v) rmax = fmaxf(rmax, st[v]);
  rmax = fmaxf(rmax, __shfl_xor(rmax, 16, 32));  // combine key halves
  const float mnew  = fmaxf(mrow, rmax);
  const float scale = __expf(mrow - mnew);       // exp(-inf)=0 on first tile
  mrow = mnew;
  float rsum = 0.f;
#pragma unroll
  for (int v = 0; v < 8; ++v) {
    const float p = __expf(st[v] - mnew);
    st[v] = p;
    rsum += p;
  }
  rsum += __shfl_xor(rsum, 16, 32);
  lrow = lrow * scale + rsum;
#pragma unroll
  for (int j = 0; j < 4; ++j)
#pragma unroll
    for (int v = 0; v < 8; ++v) acc[j][v] *= scale;

  // ---- ctx^T += V^T(64x16) x P^T(16x16): 4 K-chunks x 4 d-tiles ----
#pragma unroll
  for (int kc = 0; kc < 4; ++kc) {
    // Build P^T B-fragment for this chunk: uniform half-swap of C-layout regs
    v2f pb;
    if (kc < 2) {  // contraction rows 4kc+2h+{0,1} live in regs 0..7
      const float o0 = __shfl_xor(st[4 * kc + 2], 16, 32);
      const float o1 = __shfl_xor(st[4 * kc + 3], 16, 32);
      pb.x = h ? o0 : st[4 * kc + 0];
      pb.y = h ? o1 : st[4 * kc + 1];
    } else {       // contraction rows 8..15 live in regs 0..7 of the other half
      const int u  = kc - 2;
      const float o0 = __shfl_xor(st[4 * u + 0], 16, 32);
      const float o1 = __shfl_xor(st[4 * u + 1], 16, 32);
      pb.x = h ? st[4 * u + 2] : o0;
      pb.y = h ? st[4 * u + 3] : o1;
    }
#pragma unroll
    for (int j = 0; j < 4; ++j) {  // 4 tiles over D=64
      v2f va = vf[kc][j];
      if (MASKED) {                // reference zeroes padded V rows
        const int kr0 = kbase + 4 * kc + 2 * h;
        va.x = (kr0 < len) ? va.x : 0.f;
        va.y = ((kr0 + 1) < len) ? va.y : 0.f;
      }
      acc[j] = wmma4(va, pb, acc[j]);
    }
  }
}

__global__ __launch_bounds__(256, 1) void sdpa_wmma_f32_kernel(
    const float* __restrict__ Q, const float* __restrict__ K,
    const float* __restrict__ V, const int* __restrict__ lengths,
    float* __restrict__ out) {
  const int lane = threadIdx.x & 31;
  // Wave-uniform scalars in SGPRs: scalar loop control, saddr-form loads.
  const int gw    = __builtin_amdgcn_readfirstlane(blockIdx.x * 8 + (threadIdx.x >> 5));
  const int b     = gw >> 6;                 // 64 q-tiles per batch
  const int qbase = (gw & 63) << 4;
  const int len   = __builtin_amdgcn_readfirstlane(lengths[b]);

  const int m = lane & 15;                   // row-within-tile / query index
  const int h = lane >> 4;                   // lane half (K-chunk half select)

  const float* Qb = Q + (size_t)b * (S_LEN * D_DIM);
  const float* Kb = K + (size_t)b * (S_LEN * D_DIM);
  const float* Vb = V + (size_t)b * (S_LEN * D_DIM);

  const v2f zero2 = {0.f, 0.f};

  // ---- Q^T B-operand fragments (64 x 16 queries), resident in VGPRs ----
  // B layout chunk kc: lanes0-15 hold K=4kc+{0,1}, lanes16-31 hold K=4kc+{2,3}
  v2f qf[16];
  {
    const int   qrow = qbase + m;
    const bool  qv   = qrow < len;           // reference zeroes padded Q rows
    const float* qp  = Qb + (size_t)qrow * D_DIM + 2 * h;
#pragma unroll
    for (int kc = 0; kc < 16; ++kc) qf[kc] = *(const v2f*)(qp + 4 * kc);
#pragma unroll
    for (int kc = 0; kc < 16; ++kc) qf[kc] = qv ? qf[kc] : zero2;
  }

  // ctx^T accumulators: 4 tiles of 16(d) x 16(q), VGPR dim = d, lane dim = q
  v8f acc[4];
#pragma unroll
  for (int j = 0; j < 4; ++j)
#pragma unroll
    for (int v = 0; v < 8; ++v) acc[j][v] = 0.f;

  float mrow = -INFINITY;  // per-query running max (lane-resident, both halves)
  float lrow = 0.f;        // per-query running denominator

  // ---- steady state: fully-valid key tiles, zero masking overhead ----
  const int nfull = len >> 4;
  for (int t = 0; t < nfull; ++t)
    process_tile<false>(Kb, Vb, qf, acc, mrow, lrow, t << 4, len, m, h);

  // ---- at most one partial tile straddling len ----
  int ntiles = nfull;
  if (len & 15) {
    process_tile<true>(Kb, Vb, qf, acc, mrow, lrow, nfull << 4, len, m, h);
    ++ntiles;
  }

  // ---- analytic tail: fully-masked key tiles each add weight exp(1.25e-11) ----
  const int nrem = S_LEN - (ntiles << 4);
  if (nrem > 0) {
    const float c     = 1e-10f * 0.125f;
    const float mnew  = fmaxf(mrow, c);
    const float scale = __expf(mrow - mnew);
    const float p     = __expf(c - mnew);
    lrow = lrow * scale + (float)nrem * p;
    mrow = mnew;
#pragma unroll
    for (int j = 0; j < 4; ++j)
#pragma unroll
      for (int v = 0; v < 8; ++v) acc[j][v] *= scale;
  }

  // ---- normalize and store: lane writes query row qbase+m, d = 16j+8h+v ----
  const float inv  = 1.0f / lrow;
  const int   qrow = qbase + m;
  float* op = out + ((size_t)b * S_LEN + qrow) * D_DIM + 8 * h;
#pragma unroll
  for (int j = 0; j < 4; ++j) {
    float4 lo4 = make_float4(acc[j][0] * inv, acc[j][1] * inv,
                             acc[j][2] * inv, acc[j][3] * inv);
    float4 hi4 = make_float4(acc[j][4] * inv, acc[j][5] * inv,
                             acc[j][6] * inv, acc[j][7] * inv);
    *(float4*)(op + 16 * j)     = lo4;
    *(float4*)(op + 16 * j + 4) = hi4;
  }
}

extern "C" void kernel_launch(void* const* d_in, const int* in_sizes, int n_in,
                              void* d_out, int out_size, void* d_ws,
                              size_t ws_size, hipStream_t stream) {
  (void)in_sizes; (void)n_in; (void)out_size; (void)d_ws; (void)ws_size;
  const float* Q       = (const float*)d_in[0];
  const float* K       = (const float*)d_in[1];
  const float* V       = (const float*)d_in[2];
  const int*   lengths = (const int*)d_in[3];
  float*       out     = (float*)d_out;

  // 32 batches * 64 query-tiles = 2048 waves; 8 waves (256 threads) per block,
  // all 8 waves of a block share one batch's K/V (L1/L2 reuse).
  dim3 grid(256), block(256);
  sdpa_wmma_f32_kernel<<<grid, block, 0, stream>>>(Q, K, V, lengths, out);
}